// GIN_Encoder_56599079026907
// MI455X (gfx1250) — compile-verified
//
#include <hip/hip_runtime.h>

#define NFEAT   128
#define NHID    64
#define NLAYER  3
#define NGRAPHS 64
#define BN_EPS  1e-5f

typedef __attribute__((ext_vector_type(2))) float v2f;
typedef __attribute__((ext_vector_type(8))) float v8f;

// ---------------------------------------------------------------- utilities
__global__ __launch_bounds__(256) void fill_zero(float* __restrict__ p, int n) {
  int i = blockIdx.x * 256 + threadIdx.x;
  if (i < n) p[i] = 0.0f;
}

__global__ __launch_bounds__(256) void copy_buf4(const float4* __restrict__ s,
                                                 float4* __restrict__ d, int n4) {
  int i = blockIdx.x * 256 + threadIdx.x;
  if (i < n4) d[i] = s[i];
}

__global__ __launch_bounds__(256) void count_nodes(const int* __restrict__ batch,
                                                   float* __restrict__ cnt, int M) {
  int i = blockIdx.x * 256 + threadIdx.x;
  if (i < M) atomicAdd(&cnt[batch[i]], 1.0f);
}

// -------------------------------------------------- WMMA GEMM: C = act(A·W + b)
// A: [M,K] row-major, W: [K,64] row-major, C: [M,64].
// One wave per 16-row strip; 4 f32 16x16 accumulators tile N=64.
// W staged in LDS as paired float2 (W[2p][c], W[2p+1][c]) so every B fragment
// is one aligned ds_load_b64 straight into the WMMA source register pair.
template <int K, bool RELU>
__global__ __launch_bounds__(256) void gemm_wmma(const float* __restrict__ A,
                                                 const float* __restrict__ W,
                                                 const float* __restrict__ bias,
                                                 float* __restrict__ C, int M) {
  __shared__ v2f sW[(K / 2) * NHID];
  for (int i = threadIdx.x; i < (K / 2) * NHID; i += 256) {
    int p = i >> 6;
    int col = i & 63;
    v2f w;
    w.x = W[(2 * p) * NHID + col];
    w.y = W[(2 * p + 1) * NHID + col];
    sW[i] = w;
  }
  __syncthreads();

  const int wave = threadIdx.x >> 5;
  const int lane = threadIdx.x & 31;
  const int half = lane >> 4;   // 0: lanes 0-15, 1: lanes 16-31
  const int r    = lane & 15;
  const int m0   = (blockIdx.x * 8 + wave) * 16;
  if (m0 >= M) return;

  const float* arow = A + (size_t)(m0 + r) * K;

  v8f acc0 = {}; v8f acc1 = {}; v8f acc2 = {}; v8f acc3 = {};

  #pragma unroll 4
  for (int k = 0; k < K; k += 4) {
    const int ko = k + 2 * half;       // A 16x4 f32 layout: half selects the K pair
    const int p  = ko >> 1;            // paired-W row index

    v2f a = *(const v2f*)(arow + ko);  // 8B aligned: ko even, rows K-aligned

    const v2f* bp = sW + p * NHID + r;
    v2f b0 = bp[0];
    v2f b1 = bp[16];
    v2f b2 = bp[32];
    v2f b3 = bp[48];

    acc0 = __builtin_amdgcn_wmma_f32_16x16x4_f32(false, a, false, b0, (short)0, acc0, false, false);
    acc1 = __builtin_amdgcn_wmma_f32_16x16x4_f32(false, a, false, b1, (short)0, acc1, false, false);
    acc2 = __builtin_amdgcn_wmma_f32_16x16x4_f32(false, a, false, b2, (short)0, acc2, false, false);
    acc3 = __builtin_amdgcn_wmma_f32_16x16x4_f32(false, a, false, b3, (short)0, acc3, false, false);
  }

  const float bv0 = bias[r];
  const float bv1 = bias[16 + r];
  const float bv2 = bias[32 + r];
  const float bv3 = bias[48 + r];

  // C/D layout: vgpr j, half h -> row m0 + j + 8*h ; col = ntile*16 + r
  #pragma unroll
  for (int j = 0; j < 8; ++j) {
    float* cr = C + (size_t)(m0 + j + 8 * half) * NHID;
    float v0 = acc0[j] + bv0;
    float v1 = acc1[j] + bv1;
    float v2 = acc2[j] + bv2;
    float v3 = acc3[j] + bv3;
    if (RELU) {
      v0 = fmaxf(v0, 0.0f); v1 = fmaxf(v1, 0.0f);
      v2 = fmaxf(v2, 0.0f); v3 = fmaxf(v3, 0.0f);
    }
    cr[r]      = v0;
    cr[16 + r] = v1;
    cr[32 + r] = v2;
    cr[48 + r] = v3;
  }
}

// -------------------------------------------------- edge scatter: agg[dst] += h[src]
// 4 channels/thread: b128 gather, 4 f32 atomics; 16 threads share one edge.
__global__ __launch_bounds__(256) void edge_agg(const int* __restrict__ src,
                                                const int* __restrict__ dst,
                                                const float* __restrict__ h,
                                                float* __restrict__ agg, int E) {
  int t = blockIdx.x * 256 + threadIdx.x;
  int e  = t >> 4;
  int c4 = (t & 15) * 4;
  if (e < E) {
    const float4 v = *(const float4*)(h + (size_t)src[e] * NHID + c4);
    float* a = agg + (size_t)dst[e] * NHID + c4;
    atomicAdd(a + 0, v.x);
    atomicAdd(a + 1, v.y);
    atomicAdd(a + 2, v.z);
    atomicAdd(a + 3, v.w);
  }
}

// -------------------------------------------------- per-channel sum / sumsq
__global__ __launch_bounds__(256) void colstats(const float* __restrict__ X, int M,
                                                float* __restrict__ stats) {
  const int c  = threadIdx.x & 63;
  const int rg = threadIdx.x >> 6;  // 0..3
  const int base = blockIdx.x * 128;
  float s = 0.0f, s2 = 0.0f;
  for (int rr = rg; rr < 128; rr += 4) {
    int row = base + rr;
    if (row < M) {
      float v = X[(size_t)row * NHID + c];
      s += v;
      s2 += v * v;
    }
  }
  __shared__ float red[2][4][64];
  red[0][rg][c] = s;
  red[1][rg][c] = s2;
  __syncthreads();
  if (rg == 0) {
    s  = red[0][0][c] + red[0][1][c] + red[0][2][c] + red[0][3][c];
    s2 = red[1][0][c] + red[1][1][c] + red[1][2][c] + red[1][3][c];
    atomicAdd(&stats[c], s);
    atomicAdd(&stats[64 + c], s2);
  }
}

// -------------------------------------------------- BN params -> per-channel scale/shift
__global__ __launch_bounds__(64) void bnparam(const float* __restrict__ stats,
                                              const float* __restrict__ gamma,
                                              const float* __restrict__ beta,
                                              float* __restrict__ ss, int M) {
  int c = threadIdx.x;
  if (c < NHID) {
    float invM  = 1.0f / (float)M;
    float mean  = stats[c] * invM;
    float var   = stats[64 + c] * invM - mean * mean;
    float scale = gamma[c] * rsqrtf(var + BN_EPS);
    ss[c]       = scale;
    ss[64 + c]  = beta[c] - mean * scale;
  }
}

// -------------------------------------------------- BN apply + segment-sum pooling
// Block covers 16 consecutive nodes (4 channels/thread, b128 I/O). batch is
// sorted, so blocks are almost always graph-uniform: reduce in LDS (ds_add_f32)
// and emit 64 global atomics/block instead of 1024. Fallback at graph seams.
__global__ __launch_bounds__(256) void bn_apply_pool(const float* __restrict__ hpre,
                                                     const float* __restrict__ ss,
                                                     const int* __restrict__ batch,
                                                     float* __restrict__ h,
                                                     float* __restrict__ pool, int M) {
  __shared__ float acc[64];
  __shared__ int uniform;

  const int n0 = blockIdx.x * 16;
  if (threadIdx.x < 64) acc[threadIdx.x] = 0.0f;
  if (threadIdx.x == 0) {
    int nLast = n0 + 15 < M ? n0 + 15 : M - 1;
    uniform = (batch[n0] == batch[nLast]) ? 1 : 0;
  }
  __syncthreads();

  const int node = n0 + (threadIdx.x >> 4);
  const int c4   = (threadIdx.x & 15) * 4;
  const int un   = uniform;

  if (node < M) {
    float4 hp = *(const float4*)(hpre + (size_t)node * NHID + c4);
    float4 v;
    v.x = hp.x * ss[c4 + 0] + ss[64 + c4 + 0];
    v.y = hp.y * ss[c4 + 1] + ss[64 + c4 + 1];
    v.z = hp.z * ss[c4 + 2] + ss[64 + c4 + 2];
    v.w = hp.w * ss[c4 + 3] + ss[64 + c4 + 3];
    *(float4*)(h + (size_t)node * NHID + c4) = v;

    if (un) {
      atomicAdd(&acc[c4 + 0], v.x);
      atomicAdd(&acc[c4 + 1], v.y);
      atomicAdd(&acc[c4 + 2], v.z);
      atomicAdd(&acc[c4 + 3], v.w);
    } else {
      float* pg = pool + (size_t)batch[node] * NHID + c4;
      atomicAdd(pg + 0, v.x);
      atomicAdd(pg + 1, v.y);
      atomicAdd(pg + 2, v.z);
      atomicAdd(pg + 3, v.w);
    }
  }
  __syncthreads();
  if (un && threadIdx.x < 64) {
    atomicAdd(&pool[(size_t)batch[n0] * NHID + threadIdx.x], acc[threadIdx.x]);
  }
}

// -------------------------------------------------- pooled mean -> output accumulate
__global__ __launch_bounds__(256) void pool_out(const float* __restrict__ pool,
                                                const float* __restrict__ cnt,
                                                float* __restrict__ out, int accum) {
  int t = blockIdx.x * 256 + threadIdx.x;
  if (t < NGRAPHS * NHID) {
    int g = t >> 6;
    float v = pool[t] / fmaxf(cnt[g], 1.0f);
    out[t] = accum ? (out[t] + v) : v;
  }
}

// ---------------------------------------------------------------- driver
extern "C" void kernel_launch(void* const* d_in, const int* in_sizes, int n_in,
                              void* d_out, int out_size, void* d_ws, size_t ws_size,
                              hipStream_t stream) {
  const float* x     = (const float*)d_in[0];
  const int*   ei    = (const int*)d_in[1];
  const int*   batch = (const int*)d_in[2];
  const float* W_t   = (const float*)d_in[3];
  const float* b_t   = (const float*)d_in[4];
  const float* bn0_g = (const float*)d_in[5];
  const float* bn0_b = (const float*)d_in[6];
  const float* W1    = (const float*)d_in[7];
  const float* b1    = (const float*)d_in[8];
  const float* W2    = (const float*)d_in[9];
  const float* b2    = (const float*)d_in[10];
  const float* bn_g  = (const float*)d_in[11];
  const float* bn_b  = (const float*)d_in[12];

  const int M = in_sizes[0] / NFEAT;   // 100000
  const int E = in_sizes[1] / 2;       // 1600000
  const int* esrc = ei;
  const int* edst = ei + E;

  float* ws = (float*)d_ws;
  const size_t nodeElems = (size_t)M * NHID;
  float* hpre  = ws;
  float* h     = hpre + nodeElems;
  float* agg   = h + nodeElems;
  float* zr    = agg + nodeElems;
  float* stats = zr + nodeElems;   // 128 floats: sum, sumsq
  float* ss    = stats + 128;      // 128 floats: scale, shift
  float* cnt   = ss + 128;         // 64
  float* pool  = cnt + 64;         // 64*64
  float* out   = (float*)d_out;

  const int strips     = (M + 15) / 16;
  const int gemmBlocks = (strips + 7) / 8;
  const int poolBlocks = (M + 15) / 16;
  const int copyBlocks = (int)((nodeElems / 4 + 255) / 256);
  const int edgeBlks   = (int)(((long long)E * 16 + 255) / 256);

  // graph counts
  fill_zero<<<1, 256, 0, stream>>>(cnt, NGRAPHS);
  count_nodes<<<(M + 255) / 256, 256, 0, stream>>>(batch, cnt, M);

  // h_pre = x @ W_t + b_t
  gemm_wmma<NFEAT, false><<<gemmBlocks, 256, 0, stream>>>(x, W_t, b_t, hpre, M);

  // BN0 + pool layer 0
  fill_zero<<<1, 256, 0, stream>>>(stats, 128);
  colstats<<<(M + 127) / 128, 256, 0, stream>>>(hpre, M, stats);
  bnparam<<<1, 64, 0, stream>>>(stats, bn0_g, bn0_b, ss, M);
  fill_zero<<<16, 256, 0, stream>>>(pool, NGRAPHS * NHID);
  bn_apply_pool<<<poolBlocks, 256, 0, stream>>>(hpre, ss, batch, h, pool, M);
  pool_out<<<16, 256, 0, stream>>>(pool, cnt, out, 0);

  for (int l = 0; l < NLAYER; ++l) {
    // agg := h, then scatter-add h[src] -> agg[dst]  (agg becomes z = h + Σ)
    copy_buf4<<<copyBlocks, 256, 0, stream>>>((const float4*)h, (float4*)agg,
                                              (int)(nodeElems / 4));
    edge_agg<<<edgeBlks, 256, 0, stream>>>(esrc, edst, h, agg, E);

    // zr = relu(z @ W1 + b1) ; hpre = relu(zr @ W2 + b2)
    gemm_wmma<NHID, true><<<gemmBlocks, 256, 0, stream>>>(agg, W1 + l * NHID * NHID,
                                                          b1 + l * NHID, zr, M);
    gemm_wmma<NHID, true><<<gemmBlocks, 256, 0, stream>>>(zr, W2 + l * NHID * NHID,
                                                          b2 + l * NHID, hpre, M);

    // BN + pool
    fill_zero<<<1, 256, 0, stream>>>(stats, 128);
    colstats<<<(M + 127) / 128, 256, 0, stream>>>(hpre, M, stats);
    bnparam<<<1, 64, 0, stream>>>(stats, bn_g + l * NHID, bn_b + l * NHID, ss, M);
    fill_zero<<<16, 256, 0, stream>>>(pool, NGRAPHS * NHID);
    bn_apply_pool<<<poolBlocks, 256, 0, stream>>>(hpre, ss, batch, h, pool, M);
    pool_out<<<16, 256, 0, stream>>>(pool, cnt, out, 1);
  }
}